// Quantizer_69896297775277
// MI455X (gfx1250) — compile-verified
//
#include <hip/hip_runtime.h>
#include <hip/hip_bf16.h>

typedef __attribute__((ext_vector_type(2))) float v2f;
typedef __attribute__((ext_vector_type(8))) float v8f;

#define KCODES 1024
#define CDIM   64
#define HW     4096          // H*W = 64*64
#define NVEC   131072        // B*H*W
#define KT     128           // codes per LDS tile
#define KTP    (KT + 1)      // padded LDS stride (bank = (c+k)%64)
#define MTILE  256           // rows per block (8 waves x 32 rows)

// Workspace layout (floats): [0..1023] counts, [1024] loss accum, [1025..2048] code norms
#define WS_LOSS  1024
#define WS_NORMS 1025

__global__ __launch_bounds__(1024) void vq_prep(const float* __restrict__ cb,
                                                float* __restrict__ ws) {
    int k = threadIdx.x;            // 0..1023
    float s = 0.0f;
    #pragma unroll 8
    for (int c = 0; c < CDIM; ++c) {
        float v = cb[k * CDIM + c];
        s += v * v;
    }
    ws[k] = 0.0f;                   // counts
    ws[WS_NORMS + k] = s;           // ||e_k||^2
    if (k == 0) ws[WS_LOSS] = 0.0f; // loss accumulator
}

__global__ __launch_bounds__(256) void vq_main(const float* __restrict__ x,
                                               const float* __restrict__ cb,
                                               float* __restrict__ qout,
                                               float* __restrict__ ws) {
    __shared__ float ldsB[CDIM][KTP];   // codebook tile, c-major, padded
    __shared__ int   ldsIdx[MTILE];     // chosen code per row

    const int tid  = threadIdx.x;
    const int wave = tid >> 5;
    const int lane = tid & 31;
    const int hf   = lane >> 4;         // which 16-lane half
    const int l16  = lane & 15;

    const int rowBase = blockIdx.x * MTILE + wave * 32;  // 32 rows per wave (2 A tiles)
    const int bb = rowBase / HW;                         // batch index (rows never straddle b)
    const int hw = rowBase % HW;
    const float* xbase = x + (size_t)bb * CDIM * HW + hw;  // x(row r, chan c) = xbase[c*HW + r]

    // ---- A fragments: 2 tiles x 16 rows x 64 channels, ISA 16x4 f32 A layout ----
    // step s covers channels 4s..4s+3; lane half picks K offset +2; M = l16
    v2f af0[16], af1[16];
    #pragma unroll
    for (int s = 0; s < 16; ++s) {
        int c0 = 4 * s + 2 * hf;
        af0[s].x = xbase[(size_t)(c0 + 0) * HW + l16];
        af0[s].y = xbase[(size_t)(c0 + 1) * HW + l16];
        af1[s].x = xbase[(size_t)(c0 + 0) * HW + 16 + l16];
        af1[s].y = xbase[(size_t)(c0 + 1) * HW + 16 + l16];
    }

    float mv0[8], mv1[8];
    int   mi0[8], mi1[8];
    #pragma unroll
    for (int j = 0; j < 8; ++j) {
        mv0[j] = 3.4e38f; mi0[j] = 0;
        mv1[j] = 3.4e38f; mi1[j] = 0;
    }

    const float* __restrict__ norms = ws + WS_NORMS;

    for (int t = 0; t < KCODES / KT; ++t) {
        __syncthreads();
        // stage codebook tile transposed: ldsB[c][k] = cb[(t*KT+k)*64 + c]
        for (int i = tid; i < KT * CDIM; i += 256) {
            int k = i >> 6;      // 0..KT-1
            int c = i & 63;
            ldsB[c][k] = cb[(size_t)(t * KT + k) * CDIM + c];
        }
        __syncthreads();

        for (int kb = 0; kb < KT; kb += 16) {
            // ---- load phase: 16 pipelined DS loads into distinct registers ----
            v2f bf[16];
            #pragma unroll
            for (int s = 0; s < 16; ++s) {
                int c0 = 4 * s + 2 * hf;
                bf[s].x = ldsB[c0 + 0][kb + l16];
                bf[s].y = ldsB[c0 + 1][kb + l16];
            }
            // ---- compute phase: 32 WMMAs, two independent accumulation chains ----
            v8f acc0 = {0.f, 0.f, 0.f, 0.f, 0.f, 0.f, 0.f, 0.f};
            v8f acc1 = {0.f, 0.f, 0.f, 0.f, 0.f, 0.f, 0.f, 0.f};
            #pragma unroll
            for (int s = 0; s < 16; ++s) {
                acc0 = __builtin_amdgcn_wmma_f32_16x16x4_f32(
                    false, af0[s], false, bf[s], (short)0, acc0, false, false);
                acc1 = __builtin_amdgcn_wmma_f32_16x16x4_f32(
                    false, af1[s], false, bf[s], (short)0, acc1, false, false);
            }
            // acc[j]: row M = j + 8*hf (within tile), code N = tile base + l16
            int code = t * KT + kb + l16;
            float nrm = norms[code];
            #pragma unroll
            for (int j = 0; j < 8; ++j) {
                float d0 = nrm - 2.0f * acc0[j];    // ||e||^2 - 2 x.e  (||x||^2 dropped)
                float d1 = nrm - 2.0f * acc1[j];
                if (d0 < mv0[j]) { mv0[j] = d0; mi0[j] = code; }  // strict < keeps lowest idx
                if (d1 < mv1[j]) { mv1[j] = d1; mi1[j] = code; }
            }
        }
    }

    // ---- argmin across the 16 lanes of each half (codes are lane-striped) ----
    #pragma unroll
    for (int j = 0; j < 8; ++j) {
        float v0 = mv0[j]; int i0 = mi0[j];
        float v1 = mv1[j]; int i1 = mi1[j];
        #pragma unroll
        for (int off = 1; off < 16; off <<= 1) {
            float w0 = __shfl_xor(v0, off, 32);
            int   k0 = __shfl_xor(i0, off, 32);
            float w1 = __shfl_xor(v1, off, 32);
            int   k1 = __shfl_xor(i1, off, 32);
            if (w0 < v0 || (w0 == v0 && k0 < i0)) { v0 = w0; i0 = k0; }
            if (w1 < v1 || (w1 == v1 && k1 < i1)) { v1 = w1; i1 = k1; }
        }
        mi0[j] = i0;
        mi1[j] = i1;
    }
    if (l16 == 0) {
        #pragma unroll
        for (int j = 0; j < 8; ++j) {
            ldsIdx[wave * 32 + j + 8 * hf]      = mi0[j];
            ldsIdx[wave * 32 + 16 + j + 8 * hf] = mi1[j];
        }
    }
    __syncthreads();

    // histogram: one atomic per row (32 rows per wave, one per lane)
    atomicAdd(ws + ldsIdx[wave * 32 + hf * 16 + l16], 1.0f);

    // ---- gather quantized, write q_out [B,C,H,W], accumulate loss ----
    float* __restrict__ obase = qout + (size_t)bb * CDIM * HW + hw;
    float lsum = 0.0f;
    #pragma unroll
    for (int g = 0; g < 2; ++g) {
        const int r   = g * 16 + l16;
        const int idx = ldsIdx[wave * 32 + r];
        const float* __restrict__ qrow = cb + (size_t)idx * CDIM;
        #pragma unroll 4
        for (int it = 0; it < 32; ++it) {
            int c = 2 * it + hf;             // both halves together cover c = 0..63
            float q  = qrow[c];
            float xv = xbase[(size_t)c * HW + r];
            float dq = q - xv;
            lsum += dq * dq;
            obase[(size_t)c * HW + r] = q;
        }
    }
    #pragma unroll
    for (int off = 1; off < 32; off <<= 1) lsum += __shfl_xor(lsum, off, 32);
    if (lane == 0) atomicAdd(ws + WS_LOSS, lsum);
}

__global__ __launch_bounds__(1024) void vq_finalize(const float* __restrict__ ws,
                                                    const float* __restrict__ beta,
                                                    float* __restrict__ out,
                                                    int out_size) {
    __shared__ float partial[32];
    int tid = threadIdx.x;                   // 0..1023 == code id
    float e = ws[tid] * (1.0f / (float)NVEC);
    float term = e * logf(e + 1e-10f);

    float s = term;
    #pragma unroll
    for (int off = 1; off < 32; off <<= 1) s += __shfl_xor(s, off, 32);
    if ((tid & 31) == 0) partial[tid >> 5] = s;
    __syncthreads();
    if (tid < 32) {
        float t2 = partial[tid];
        #pragma unroll
        for (int off = 1; off < 32; off <<= 1) t2 += __shfl_xor(t2, off, 32);
        if (tid == 0) {
            out[out_size - 1] = expf(-t2);                       // perplexity
            float b = beta[0];
            out[0] = ws[WS_LOSS] * (1.0f + b) / (float)(NVEC * (size_t)CDIM);  // loss
        }
    }
}

extern "C" void kernel_launch(void* const* d_in, const int* in_sizes, int n_in,
                              void* d_out, int out_size, void* d_ws, size_t ws_size,
                              hipStream_t stream) {
    const float* x    = (const float*)d_in[0];   // [32,64,64,64]
    const float* cb   = (const float*)d_in[1];   // [1024,64]
    const float* beta = (const float*)d_in[2];   // scalar
    float* out = (float*)d_out;                  // [loss, q_out(BCHW), perplexity]
    float* ws  = (float*)d_ws;                   // counts | loss | norms  (~8.2 KB)

    vq_prep<<<1, 1024, 0, stream>>>(cb, ws);
    vq_main<<<NVEC / MTILE, 256, 0, stream>>>(x, cb, out + 1, ws);
    vq_finalize<<<1, 1024, 0, stream>>>(ws, beta, out, out_size);
}